// volterraModule_240518168865
// MI455X (gfx1250) — compile-verified
//
#include <hip/hip_runtime.h>

// MI455X / gfx1250, wave32. fp32 WMMA (V_WMMA_F32_16X16X4_F32) for all dense
// GEMM parts; LDS-staged factored contractions for the Kronecker heads.

typedef float v2f __attribute__((ext_vector_type(2)));
typedef float v8f __attribute__((ext_vector_type(8)));

#define NB    16    // batch
#define LDIM  12
#define LW    144   // 12*12 flattened input per batch
#define RR    16    // tucker rank
#define NPAIR 78    // #{(i1,i2): i1<=i2<12}
#define NTRI  364   // #{(j1,j2,j3): j1<=j2<=j3<12}
#define NOUT  448

__device__ __forceinline__ int pairOff(int i1) {
    // offset of first pair (i1, i1) in the compact i1<=i2 enumeration
    return i1 * 12 - (i1 * (i1 - 1)) / 2;
}

// ---------------------------------------------------------------------------
// Order-2 head: t[b,p,q] = sum_{i1<=i2, j1<=j2} U[(i1,i2),p] V[(j1,j2),q]
//                                               x[b,i1,j1] x[b,i2,j2]
// One block per batch. LDS: x tile + S[i1,p,j2] = sum_{i2>=i1} U*x.
// ---------------------------------------------------------------------------
__global__ __launch_bounds__(256) void order2_kernel(
    const float* __restrict__ x,      // [16,12,12]
    const float* __restrict__ U,      // [144,16]
    const float* __restrict__ V,      // [144,16]
    float* __restrict__ tout)         // [16,256]
{
    __shared__ float xs[LW];
    __shared__ float S[12 * RR * 12]; // (i1,p,j2) -> 2304 floats
    const int b = blockIdx.x;
    const int tid = threadIdx.x;

    if (tid < LW) xs[tid] = x[b * LW + tid];
    __syncthreads();

    // Stage 1: S[(i1*16+p)*12 + j2] = sum_{i2>=i1} U[(i1*12+i2)*16+p] * xs[i2*12+j2]
    for (int e = tid; e < 12 * RR * 12; e += 256) {
        int j2 = e % 12;
        int t  = e / 12;
        int p  = t % RR;
        int i1 = t / RR;
        float s = 0.f;
        for (int i2 = i1; i2 < 12; ++i2)
            s += U[(i1 * 12 + i2) * RR + p] * xs[i2 * 12 + j2];
        S[e] = s;
    }
    __syncthreads();

    // Stage 2: fold in masked V side. thread = (p,q).
    const int p = tid >> 4;
    const int q = tid & 15;
    float acc = 0.f;
    for (int j1 = 0; j1 < 12; ++j1) {
        for (int j2 = j1; j2 < 12; ++j2) {
            float inner = 0.f;
            for (int i1 = 0; i1 < 12; ++i1)
                inner += xs[i1 * 12 + j1] * S[(i1 * RR + p) * 12 + j2];
            acc += V[(j1 * 12 + j2) * RR + q] * inner;
        }
    }
    tout[b * 256 + p * RR + q] = acc;
}

// ---------------------------------------------------------------------------
// Order-3 head: t[b,p,q] = sum_{i1<=i2<=i3, j1<=j2<=j3}
//     U3[(i1,i2,i3),p] V3[(j1,j2,j3),q] x[i1,j1] x[i2,j2] x[i3,j3]
// One block per batch. Two p-halves to keep static LDS < 64KB:
//   T1[(i1<=i2 pair)*8+pp, j3] = sum_{i3>=i2} U3 * x        (7488 f per half)
//   T3[p, compactJ]            = sum_{i1} x * sum_{i2>=i1} x * T1  (5824 f)
// then contract T3 against masked V3.
// ---------------------------------------------------------------------------
__global__ __launch_bounds__(256) void order3_kernel(
    const float* __restrict__ x,      // [16,12,12]
    const float* __restrict__ U3,     // [1728,16]
    const float* __restrict__ V3,     // [1728,16]
    float* __restrict__ tout)         // [16,256]
{
    __shared__ float xs[LW];
    __shared__ float T1[NPAIR * 8 * 12]; // [(pi*8+pp)*12 + j3]
    __shared__ float T3[RR * NTRI];      // [p*364 + compactJ]
    const int b = blockIdx.x;
    const int tid = threadIdx.x;

    if (tid < LW) xs[tid] = x[b * LW + tid];
    __syncthreads();

    for (int ph = 0; ph < 2; ++ph) {
        // Stage 1: T1 for p in [8*ph, 8*ph+8)
        for (int e = tid; e < NPAIR * 8 * 12; e += 256) {
            int j3 = e % 12;
            int t  = e / 12;
            int pp = t % 8;
            int pi = t / 8;
            int i1 = 0;
            while (pairOff(i1 + 1) <= pi) ++i1;
            int i2 = i1 + (pi - pairOff(i1));
            int p  = ph * 8 + pp;
            float s = 0.f;
            for (int i3 = i2; i3 < 12; ++i3)
                s += U3[(i1 * 144 + i2 * 12 + i3) * RR + p] * xs[i3 * 12 + j3];
            T1[e] = s;
        }
        __syncthreads();

        // Stage 2+3: T3[p, J] over kept triples J=(j1<=j2<=j3)
        const int p_loc  = tid >> 5;  // 8 p values, 32 threads each
        const int lane32 = tid & 31;
        int k = 0;
        for (int j1 = 0; j1 < 12; ++j1)
            for (int j2 = j1; j2 < 12; ++j2)
                for (int j3 = j2; j3 < 12; ++j3) {
                    if ((k & 31) == lane32) {
                        float s = 0.f;
                        for (int i1 = 0; i1 < 12; ++i1) {
                            float inner = 0.f;
                            for (int i2 = i1; i2 < 12; ++i2) {
                                int pi = pairOff(i1) + (i2 - i1);
                                inner += xs[i2 * 12 + j2] *
                                         T1[(pi * 8 + p_loc) * 12 + j3];
                            }
                            s += xs[i1 * 12 + j1] * inner;
                        }
                        T3[(ph * 8 + p_loc) * NTRI + k] = s;
                    }
                    ++k;
                }
        __syncthreads(); // also protects T1 reuse in next p-half
    }

    // Stage 4: t[p,q] = sum_J T3[p,J] * V3[J,q]
    const int p = tid >> 4;
    const int q = tid & 15;
    float acc = 0.f;
    int k = 0;
    for (int j1 = 0; j1 < 12; ++j1)
        for (int j2 = j1; j2 < 12; ++j2)
            for (int j3 = j2; j3 < 12; ++j3) {
                acc += T3[p * NTRI + k] * V3[(j1 * 144 + j2 * 12 + j3) * RR + q];
                ++k;
            }
    tout[b * 256 + p * RR + q] = acc;
}

// ---------------------------------------------------------------------------
// Final pass: out[b, n] for all 448 columns via fp32 WMMA (16x16x4).
// One wave per 16-column tile. A rows = batch (M=16), B = weight/core slabs.
// Fragment layouts per CDNA5 ISA 7.12.2:
//   A 16x4 f32: v[0]={K0 | K2}, v[1]={K1 | K3} across lane halves, M=lane&15
//   B 4x16 f32: mirrored, N=lane&15
//   C/D:        v[r] -> row M=r (lanes 0-15) / M=r+8 (lanes 16-31), N=lane&15
// ---------------------------------------------------------------------------
__global__ __launch_bounds__(32) void out_wmma_kernel(
    const float* __restrict__ X,                          // [16,144]
    const float* __restrict__ K0, const float* __restrict__ b0,
    const float* __restrict__ K1, const float* __restrict__ b1,
    const float* __restrict__ C1,                         // [256,128]
    const float* __restrict__ K2, const float* __restrict__ b2,
    const float* __restrict__ C2, const float* __restrict__ C3, // [256,256]
    const float* __restrict__ t2a, const float* __restrict__ t2b,
    const float* __restrict__ t3,                         // [16,256] each
    float* __restrict__ out)                              // [16,1,448]
{
    const int lane = threadIdx.x;
    const int hi = lane >> 4;       // lane half
    const int mr = lane & 15;       // row (A) / col (B,D) within tile
    const int n0 = blockIdx.x * 16; // global output column of this tile

    const float* Wk; const float* bias; int ldn; int col;
    if (n0 < 64)       { Wk = K0; bias = b0; ldn = 64;  col = n0; }
    else if (n0 < 192) { Wk = K1; bias = b1; ldn = 128; col = n0 - 64; }
    else               { Wk = K2; bias = b2; ldn = 256; col = n0 - 192; }

    v8f acc = {0.f, 0.f, 0.f, 0.f, 0.f, 0.f, 0.f, 0.f};

    // order-1 GEMM: [16 x 144] * [144 x ldn] tile
    for (int k = 0; k < LW; k += 4) {
        v2f a, bf;
        a[0]  = X[mr * LW + k + 2 * hi];
        a[1]  = X[mr * LW + k + 2 * hi + 1];
        bf[0] = Wk[(k + 2 * hi) * ldn + col + mr];
        bf[1] = Wk[(k + 2 * hi + 1) * ldn + col + mr];
        acc = __builtin_amdgcn_wmma_f32_16x16x4_f32(
            false, a, false, bf, (short)0, acc, false, false);
    }

    // core contractions: [16 x 256] * [256 x ldn] tile(s)
    const float* Ts[2] = {nullptr, nullptr};
    const float* Cs[2] = {nullptr, nullptr};
    int ncore = 0;
    if (n0 >= 64 && n0 < 192) { Ts[0] = t2a; Cs[0] = C1; ncore = 1; }
    else if (n0 >= 192)       { Ts[0] = t2b; Cs[0] = C2;
                                Ts[1] = t3;  Cs[1] = C3; ncore = 2; }
    for (int c = 0; c < ncore; ++c) {
        const float* T = Ts[c];
        const float* C = Cs[c];
        for (int k = 0; k < 256; k += 4) {
            v2f a, bf;
            a[0]  = T[mr * 256 + k + 2 * hi];
            a[1]  = T[mr * 256 + k + 2 * hi + 1];
            bf[0] = C[(k + 2 * hi) * ldn + col + mr];
            bf[1] = C[(k + 2 * hi + 1) * ldn + col + mr];
            acc = __builtin_amdgcn_wmma_f32_16x16x4_f32(
                false, a, false, bf, (short)0, acc, false, false);
        }
    }

    const float bv = bias[col + mr];
#pragma unroll
    for (int v = 0; v < 8; ++v)
        out[(v + 8 * hi) * NOUT + n0 + mr] = acc[v] + bv;
}

// ---------------------------------------------------------------------------
extern "C" void kernel_launch(void* const* d_in, const int* in_sizes, int n_in,
                              void* d_out, int out_size, void* d_ws, size_t ws_size,
                              hipStream_t stream) {
    const float* x   = (const float*)d_in[0];
    const float* K0  = (const float*)d_in[1];
    const float* b0  = (const float*)d_in[2];
    const float* K1  = (const float*)d_in[3];
    const float* b1  = (const float*)d_in[4];
    const float* U2a = (const float*)d_in[5];
    const float* V2a = (const float*)d_in[6];
    const float* C1  = (const float*)d_in[7];
    const float* K2  = (const float*)d_in[8];
    const float* b2  = (const float*)d_in[9];
    const float* U2b = (const float*)d_in[10];
    const float* V2b = (const float*)d_in[11];
    const float* C2  = (const float*)d_in[12];
    const float* U3  = (const float*)d_in[13];
    const float* V3  = (const float*)d_in[14];
    const float* C3  = (const float*)d_in[15];
    float* out = (float*)d_out;

    float* ws  = (float*)d_ws;
    float* t2a = ws;                 // [16,256]
    float* t2b = ws + NB * 256;      // [16,256]
    float* t3  = ws + 2 * NB * 256;  // [16,256]

    order2_kernel<<<NB, 256, 0, stream>>>(x, U2a, V2a, t2a);
    order2_kernel<<<NB, 256, 0, stream>>>(x, U2b, V2b, t2b);
    order3_kernel<<<NB, 256, 0, stream>>>(x, U3, V3, t3);
    out_wmma_kernel<<<NOUT / 16, 32, 0, stream>>>(
        x, K0, b0, K1, b1, C1, K2, b2, C2, C3, t2a, t2b, t3, out);
}